// GraphAttention_73572789781093
// MI455X (gfx1250) — compile-verified
//
#include <hip/hip_runtime.h>
#include <math.h>

#define N_NODES 100000
#define N_EDGES 1600000
#define F_IN    128
#define F_OUT   32
#define HEADS   8
#define C_TOT   (HEADS * F_OUT)   // 256 output columns
#define LRELU_ALPHA 0.2f

typedef float v2f __attribute__((ext_vector_type(2)));
typedef float v8f __attribute__((ext_vector_type(8)));

// ---------------------------------------------------------------------------
// K1: row[] is sorted (by construction in the reference). Build CSR offsets
// with a per-node binary search -> fully deterministic, no atomics.
// ---------------------------------------------------------------------------
__global__ void build_row_ptr(const int* __restrict__ row, int* __restrict__ row_ptr) {
    int n = blockIdx.x * blockDim.x + threadIdx.x;
    if (n > N_NODES) return;
    int lo = 0, hi = N_EDGES;
    while (lo < hi) {
        int mid = (lo + hi) >> 1;
        if (row[mid] < n) lo = mid + 1; else hi = mid;
    }
    row_ptr[n] = lo;  // first edge with row[e] >= n
}

// ---------------------------------------------------------------------------
// K2: feat[N,256] = x[N,128] @ W[128,256]   (W = 8 head kernels concatenated)
// One wave computes a 16x16 f32 tile with 32x V_WMMA_F32_16X16X4_F32.
// fp32 WMMA keeps reference precision; the GEMM is HBM-bound (~154MB), so
// the fp32 matrix rate is more than sufficient.
//
// A frag (16x4 f32, ISA layout): lanes 0-15 -> K=k0,k0+1 ; lanes 16-31 -> K=k0+2,k0+3
// B frag (4x16 f32, symmetric):  lanes 0-15 -> K=k0,k0+1 ; lanes 16-31 -> K=k0+2,k0+3 ; N=lane%16
// C/D (16x16 f32): VGPR v, lanes 0-15 -> M=v ; lanes 16-31 -> M=v+8 ; N=lane%16
// ---------------------------------------------------------------------------
__global__ void __launch_bounds__(256) gat_gemm(const float* __restrict__ x,
                                                const float* __restrict__ kern,  // [8][128][32]
                                                float* __restrict__ feat) {
    const int wave = threadIdx.x >> 5;
    const int lane = threadIdx.x & 31;
    const int m0   = blockIdx.x * 16;          // 100000 / 16 = 6250 exact
    const int lr   = lane & 15;                // row-in-tile (A) / col-in-tile (B,C)
    const int kh   = lane >> 4;                // K-half selector

    // Each of the 8 waves handles two N-tiles (wave, wave+8), sharing A frags.
    const int n0a = wave * 16;
    const int n0b = (wave + 8) * 16;
    const int ca  = n0a + lr;                  // global output columns
    const int cb  = n0b + lr;
    // W[k][c] lives at kern[(c/32)*128*32 + k*32 + (c%32)]
    const float* __restrict__ Ba = kern + (size_t)(ca >> 5) * (F_IN * F_OUT) + (ca & 31);
    const float* __restrict__ Bb = kern + (size_t)(cb >> 5) * (F_IN * F_OUT) + (cb & 31);
    const float* __restrict__ arow = x + (size_t)(m0 + lr) * F_IN;

    v8f acc0 = {};
    v8f acc1 = {};
#pragma unroll 4
    for (int k0 = 0; k0 < F_IN; k0 += 4) {
        const int kk = k0 + 2 * kh;
        v2f a = *(const v2f*)(arow + kk);      // 8B, 8B-aligned
        v2f b0, b1;
        b0.x = Ba[(size_t)kk * F_OUT];
        b0.y = Ba[(size_t)(kk + 1) * F_OUT];
        b1.x = Bb[(size_t)kk * F_OUT];
        b1.y = Bb[(size_t)(kk + 1) * F_OUT];
        acc0 = __builtin_amdgcn_wmma_f32_16x16x4_f32(false, a, false, b0,
                                                     (short)0, acc0, false, false);
        acc1 = __builtin_amdgcn_wmma_f32_16x16x4_f32(false, a, false, b1,
                                                     (short)0, acc1, false, false);
    }

#pragma unroll
    for (int v = 0; v < 8; ++v) {
        const size_t r = (size_t)(m0 + v + 8 * kh) * C_TOT;
        feat[r + n0a + lr] = acc0[v];
        feat[r + n0b + lr] = acc1[v];
    }
}

// ---------------------------------------------------------------------------
// K3: per-(node,head) attention scores: a_s = feat_h . att_self_h, a_n likewise.
// ---------------------------------------------------------------------------
__global__ void att_scores(const float* __restrict__ feat,
                           const float* __restrict__ att_self,   // [8][32]
                           const float* __restrict__ att_neigh,  // [8][32]
                           float* __restrict__ a_s,
                           float* __restrict__ a_n) {
    int t = blockIdx.x * blockDim.x + threadIdx.x;
    if (t >= N_NODES * HEADS) return;
    const int n = t >> 3, h = t & 7;
    const float* __restrict__ f  = feat + (size_t)n * C_TOT + h * F_OUT;
    const float* __restrict__ ws = att_self + h * F_OUT;
    const float* __restrict__ wn = att_neigh + h * F_OUT;
    float s = 0.f, q = 0.f;
#pragma unroll
    for (int j = 0; j < F_OUT; ++j) {
        const float fv = f[j];
        s += fv * ws[j];
        q += fv * wn[j];
    }
    a_s[t] = s;
    a_n[t] = q;
}

// ---------------------------------------------------------------------------
// K4: destination-node softmax + SpMM aggregation.
// One block per node; wave h (of 8) owns head h; lane = output feature j.
// feat (102.4MB) is L2-resident on MI455X (192MB L2) -> gathers hit L2.
// Sequential per-segment accumulation => deterministic (no float atomics).
// ---------------------------------------------------------------------------
__global__ void __launch_bounds__(256) gat_aggregate(const int* __restrict__ col,
                                                     const int* __restrict__ row_ptr,
                                                     const float* __restrict__ feat,
                                                     const float* __restrict__ a_s,
                                                     const float* __restrict__ a_n,
                                                     const float* __restrict__ biases, // [8][32]
                                                     float* __restrict__ out) {
    const int n    = blockIdx.x;
    const int h    = threadIdx.x >> 5;
    const int lane = threadIdx.x & 31;
    const int start = row_ptr[n];
    const int end   = row_ptr[n + 1];

    float* __restrict__ o = out + (size_t)n * C_TOT + h * F_OUT + lane;
    const float bias = biases[h * F_OUT + lane];

    if (end == start) {        // empty segment: segment_sum == 0 -> out = bias
        *o = bias;
        return;
    }

    const float as = a_s[n * HEADS + h];

    // Pass 1: numerically-stable max over edge logits (lanes split edges).
    float m = -INFINITY;
    for (int e = start + lane; e < end; e += 32) {
        float ev = as + a_n[col[e] * HEADS + h];
        ev = ev > 0.f ? ev : LRELU_ALPHA * ev;
        m = fmaxf(m, ev);
    }
#pragma unroll
    for (int off = 16; off > 0; off >>= 1)
        m = fmaxf(m, __shfl_xor(m, off, 32));

    // Pass 2: exp/denominator + weighted feature gather (coalesced 128B/edge).
    float denom = 0.f, acc = 0.f;
    for (int e = start; e < end; ++e) {
        const int c = col[e];
        if (e + 1 < end)   // hide L2 gather latency for next edge's feature row
            __builtin_prefetch(feat + (size_t)col[e + 1] * C_TOT + h * F_OUT, 0, 0);
        float ev = as + a_n[c * HEADS + h];
        ev = ev > 0.f ? ev : LRELU_ALPHA * ev;
        const float w = __expf(ev - m);
        denom += w;
        acc += w * feat[(size_t)c * C_TOT + h * F_OUT + lane];
    }
    *o = acc / denom + bias;
}

// ---------------------------------------------------------------------------
extern "C" void kernel_launch(void* const* d_in, const int* in_sizes, int n_in,
                              void* d_out, int out_size, void* d_ws, size_t ws_size,
                              hipStream_t stream) {
    const float* x      = (const float*)d_in[0];
    const int*   row    = (const int*)  d_in[1];
    const int*   col    = (const int*)  d_in[2];
    const float* kern   = (const float*)d_in[3];
    const float* att_s  = (const float*)d_in[4];
    const float* att_n  = (const float*)d_in[5];
    const float* biases = (const float*)d_in[6];
    float* out = (float*)d_out;

    // Workspace layout (~109.3 MB): feat | a_s | a_n | row_ptr
    float* feat   = (float*)d_ws;
    float* a_s    = feat + (size_t)N_NODES * C_TOT;
    float* a_n    = a_s + (size_t)N_NODES * HEADS;
    int*   rowptr = (int*)(a_n + (size_t)N_NODES * HEADS);

    build_row_ptr<<<(N_NODES + 1 + 255) / 256, 256, 0, stream>>>(row, rowptr);
    gat_gemm<<<N_NODES / 16, 256, 0, stream>>>(x, kern, feat);
    att_scores<<<(N_NODES * HEADS + 255) / 256, 256, 0, stream>>>(feat, att_s, att_n, a_s, a_n);
    gat_aggregate<<<N_NODES, 256, 0, stream>>>(col, rowptr, feat, a_s, a_n, biases, out);

    (void)in_sizes; (void)n_in; (void)out_size; (void)ws_size;
}